// MeanShiftPP_89094801588878
// MI455X (gfx1250) — compile-verified
//
#include <hip/hip_runtime.h>
#include <hip/hip_bf16.h>
#include <stdint.h>

// Mean-shift (bandwidth 0.1, 5 steps, base-3 flat kernel) for MI455X / gfx1250.
// deposit-once + per-BIN 125-tap tent convolution + per-point single lookup
// == reference's deposit-27 / gather-27 (weights w(d)=Prod max(0,3-|d|)).

static constexpr int      BLOCK  = 256;          // 8 wave32 waves per block
static constexpr int      CAPLOG = 20;
static constexpr uint32_t CAP    = 1u << CAPLOG; // 1M hash slots (L2-resident)
static constexpr uint32_t HMASK  = CAP - 1u;
static constexpr int      EMPTYK = -1;           // valid keys are 30-bit non-negative
static constexpr float    TOL2   = 1.0e-6f;      // (1e-3)^2
static constexpr int      NSTEPS = 5;

__device__ __forceinline__ int bin_of(float v) { return (int)(v / 0.1f); } // trunc==ref
__device__ __forceinline__ int pack_key(int bx, int by, int bz) {
  return ((bx + 512) << 20) | ((by + 512) << 10) | (bz + 512);             // == ref packing
}
__device__ __forceinline__ uint32_t hash_slot(int key) {
  return ((uint32_t)key * 2654435761u) >> (32 - CAPLOG);
}

// ---- CDNA5 async global<->LDS (inline asm; ASYNCcnt-tracked) ----------------
__device__ __forceinline__ void async_ld_b128(uint32_t lds, uint64_t gaddr) {
  asm volatile("global_load_async_to_lds_b128 %0, %1, off" :: "v"(lds), "v"(gaddr) : "memory");
}
__device__ __forceinline__ void async_ld_b32(uint32_t lds, uint64_t gaddr) {
  asm volatile("global_load_async_to_lds_b32 %0, %1, off" :: "v"(lds), "v"(gaddr) : "memory");
}
__device__ __forceinline__ void async_st_b128(uint64_t gaddr, uint32_t lds) {
  asm volatile("global_store_async_from_lds_b128 %0, %1, off" :: "v"(gaddr), "v"(lds) : "memory");
}
__device__ __forceinline__ void async_st_b32(uint64_t gaddr, uint32_t lds) {
  asm volatile("global_store_async_from_lds_b32 %0, %1, off" :: "v"(gaddr), "v"(lds) : "memory");
}
__device__ __forceinline__ void wait_async0() {
  asm volatile("s_wait_asynccnt 0x0" ::: "memory");
}

// ---- kernels ---------------------------------------------------------------
__global__ void __launch_bounds__(1) ms_init(unsigned* flags) {
  flags[0] = 0u;   // done
  flags[1] = 0u;   // max displacement^2 (float bits)
}

__global__ void __launch_bounds__(BLOCK)
ms_clear(int* __restrict__ keys, float4* __restrict__ vals, unsigned* __restrict__ flags) {
  const uint32_t i = blockIdx.x * BLOCK + threadIdx.x;
  if (i == 0) flags[1] = 0u;           // reset max-disp accumulator every step
  if (flags[0] != 0u) return;          // frozen: hash unused this step
  keys[i] = EMPTYK;
  vals[i] = make_float4(0.f, 0.f, 0.f, 0.f);
}

__global__ void __launch_bounds__(BLOCK)
ms_deposit(const float* __restrict__ X, int n, int* __restrict__ keys,
           float4* __restrict__ vals, int* __restrict__ slot_of,
           const unsigned* __restrict__ flags) {
  if (flags[0] != 0u) return;
  const int p = blockIdx.x * BLOCK + threadIdx.x;
  if (p >= n) return;
  const float x = X[3 * (size_t)p + 0];
  const float y = X[3 * (size_t)p + 1];
  const float z = X[3 * (size_t)p + 2];
  const int key = pack_key(bin_of(x), bin_of(y), bin_of(z));
  uint32_t slot = hash_slot(key);
  while (true) {                        // open addressing, low load factor
    const int prev = atomicCAS(&keys[slot], EMPTYK, key);
    if (prev == EMPTYK || prev == key) break;
    slot = (slot + 1u) & HMASK;
  }
  slot_of[p] = (int)slot;               // cache: point pass needs no probing
  atomicAdd(&vals[slot].x, x);
  atomicAdd(&vals[slot].y, y);
  atomicAdd(&vals[slot].z, z);
  atomicAdd(&vals[slot].w, 1.0f);
}

// Per-occupied-bin 125-tap tent convolution of (sum,count) into U.
__global__ void __launch_bounds__(BLOCK)
ms_conv(const int* __restrict__ keys, const float4* __restrict__ vals,
        float4* __restrict__ U, const unsigned* __restrict__ flags) {
  if (flags[0] != 0u) return;
  const uint32_t i = blockIdx.x * BLOCK + threadIdx.x;
  const int key0 = keys[i];
  if (key0 == EMPTYK) return;
  const int bx = ((key0 >> 20) & 1023) - 512;
  const int by = ((key0 >> 10) & 1023) - 512;
  const int bz = ( key0        & 1023) - 512;
  float sx = 0.f, sy = 0.f, sz = 0.f, sc = 0.f;
  for (int o = 0; o < 125; ++o) {       // w(d) = 3-|d|, d in [-2,2] per dim
    const int dz = o % 5, dy = (o / 5) % 5, dx = o / 25;
    const int ax = dx - 2, ay = dy - 2, az = dz - 2;
    const float w = (float)((3 - (ax < 0 ? -ax : ax)) *
                            (3 - (ay < 0 ? -ay : ay)) *
                            (3 - (az < 0 ? -az : az)));
    const int key = pack_key(bx + ax, by + ay, bz + az);
    uint32_t slot = hash_slot(key);
    while (true) {
      const int k = keys[slot];
      if (k == key) {
        const float4 v = vals[slot];
        sx += w * v.x; sy += w * v.y; sz += w * v.z; sc += w * v.w;
        break;
      }
      if (k == EMPTYK) break;           // unoccupied neighbor: contributes 0
      slot = (slot + 1u) & HMASK;
    }
  }
  U[i] = make_float4(sx, sy, sz, sc);
}

// Per-point pass: async-stage X tile, one U lookup, divide, async write-out.
__global__ void __launch_bounds__(BLOCK)
ms_gather(const float* __restrict__ Xin, float* __restrict__ Xout, int n,
          const int* __restrict__ slot_of, const float4* __restrict__ U,
          unsigned* __restrict__ flags) {
  __shared__ __align__(16) float tile[BLOCK * 3];
  const int t   = threadIdx.x;
  const int p0  = blockIdx.x * BLOCK;
  const int cnt = min(BLOCK, n - p0);
  const int ne  = cnt * 3;

  if (flags[0] != 0u) {                 // frozen: identity pass-through
    for (int e = t; e < ne; e += BLOCK)
      Xout[(size_t)p0 * 3 + e] = Xin[(size_t)p0 * 3 + e];
    return;
  }

  // Stage the block's points coalesced via the async global->LDS pipe.
  const uint64_t gin  = (uint64_t)(uintptr_t)(Xin + (size_t)p0 * 3);
  const uint32_t lds0 = (uint32_t)(uintptr_t)tile;
  const int nchunk = (ne * 4) >> 4;                       // 16B chunks
  for (int c = t; c < nchunk; c += BLOCK) async_ld_b128(lds0 + 16u * c, gin + 16u * c);
  for (int e = nchunk * 4 + t; e < ne; e += BLOCK) async_ld_b32(lds0 + 4u * e, gin + 4u * e);
  wait_async0();
  __syncthreads();

  float d2 = 0.0f;
  if (t < cnt) {
    const float px = tile[3 * t + 0];
    const float py = tile[3 * t + 1];
    const float pz = tile[3 * t + 2];
    const float4 u = U[slot_of[p0 + t]];                  // single lookup, no probing
    const float nx = u.x / u.w, ny = u.y / u.w, nz = u.z / u.w;  // u.w >= 27
    const float ex = nx - px, ey = ny - py, ez = nz - pz;
    d2 = ex * ex + ey * ey + ez * ez;
    tile[3 * t + 0] = nx; tile[3 * t + 1] = ny; tile[3 * t + 2] = nz;
  }
  __syncthreads();

  // Write results coalesced via the async LDS->global pipe.
  const uint64_t gout = (uint64_t)(uintptr_t)(Xout + (size_t)p0 * 3);
  for (int c = t; c < nchunk; c += BLOCK) async_st_b128(gout + 16u * c, lds0 + 16u * c);
  for (int e = nchunk * 4 + t; e < ne; e += BLOCK) async_st_b32(gout + 4u * e, lds0 + 4u * e);
  wait_async0();

  // wave32 max-reduce of displacement^2, one atomic per wave (f32>=0 bits monotone).
  for (int off = 16; off > 0; off >>= 1)
    d2 = fmaxf(d2, __shfl_xor(d2, off, 32));
  if ((t & 31) == 0) atomicMax(&flags[1], __float_as_uint(d2));
}

__global__ void __launch_bounds__(1) ms_update(unsigned* flags) {
  if (__uint_as_float(flags[1]) <= TOL2) flags[0] = 1u;   // done |= conv
}

// ---- host launch -----------------------------------------------------------
extern "C" void kernel_launch(void* const* d_in, const int* in_sizes, int n_in,
                              void* d_out, int out_size, void* d_ws, size_t ws_size,
                              hipStream_t stream) {
  const float* X   = (const float*)d_in[0];
  float*       out = (float*)d_out;
  const int    n   = in_sizes[0] / 3;

  // ws: [vals CAP*16][keys CAP*4][U CAP*16][slot_of n*4][X0 n*12][X1 n*12][flags 8]
  char*     ws      = (char*)d_ws;
  float4*   vals    = (float4*)ws;
  int*      keys    = (int*)(ws + (size_t)CAP * 16);
  float4*   U       = (float4*)(ws + (size_t)CAP * 20);
  int*      slot_of = (int*)(ws + (size_t)CAP * 36);
  float*    X0      = (float*)(ws + (size_t)CAP * 36 + (size_t)n * 4);
  float*    X1      = X0 + (size_t)n * 3;
  unsigned* flags   = (unsigned*)(X1 + (size_t)n * 3);

  const int nb_pts = (n + BLOCK - 1) / BLOCK;
  const int nb_cap = (int)(CAP / BLOCK);

  ms_init<<<1, 1, 0, stream>>>(flags);

  const float* cur = X;
  float* bufs[NSTEPS] = {X0, X1, X0, X1, out};
  for (int s = 0; s < NSTEPS; ++s) {
    float* nxt = bufs[s];
    ms_clear  <<<nb_cap, BLOCK, 0, stream>>>(keys, vals, flags);
    ms_deposit<<<nb_pts, BLOCK, 0, stream>>>(cur, n, keys, vals, slot_of, flags);
    ms_conv   <<<nb_cap, BLOCK, 0, stream>>>(keys, vals, U, flags);
    ms_gather <<<nb_pts, BLOCK, 0, stream>>>(cur, nxt, n, slot_of, U, flags);
    ms_update <<<1, 1, 0, stream>>>(flags);
    cur = nxt;
  }
}